// MessagePassing_48498770706476
// MI455X (gfx1250) — compile-verified
//
#include <hip/hip_runtime.h>

// GNN message passing (gather + scatter-add), MI455X / gfx1250.
//
// out[i,:] = sum_{e : dst[e]==i} x[src[e], :]   with N=10000, E=320000, D=128 (f32)
//
// Working set (~15 MB) fits in the 192 MB L2 -> bound by L2 bandwidth and
// fp32-atomic throughput, not HBM. Strategy:
//   * one wave32 per edge: 32 lanes x float4 = 512 B = the whole feature row,
//     a single perfectly-coalesced global_load_b128 per lane
//   * edge indices forced wave-uniform (readfirstlane -> SGPR saddr addressing)
//   * scatter via native global_atomic_add_f32 (unsafeAtomicAdd)
//   * edge indices double-buffered into LDS with CDNA5 async global->LDS loads
//     (ASYNCcnt + s_wait_asynccnt) to overlap index fetch with compute.

#ifndef __has_builtin
#define __has_builtin(x) 0
#endif

#if defined(__HIP_DEVICE_COMPILE__) && \
    __has_builtin(__builtin_amdgcn_global_load_async_to_lds_b32) && \
    __has_builtin(__builtin_amdgcn_s_wait_asynccnt)
#define USE_ASYNC_LDS 1
#else
#define USE_ASYNC_LDS 0
#endif

#define WPB   8            // waves per block (wave32)
#define TPB   (WPB * 32)   // 256 threads
#define CHUNK 32           // edges staged per wave-chunk
#define D_FEAT 128

#if defined(__HIP_DEVICE_COMPILE__)
typedef __attribute__((address_space(1))) int* gptr_i32;
typedef __attribute__((address_space(3))) int* lptr_i32;
#endif

// One full wave processes one edge: lane L handles features [4L, 4L+4).
__device__ __forceinline__ void scatter_edge(const float4* __restrict__ x4,
                                             float* __restrict__ out,
                                             int s, int d, int lane) {
  // Indices are identical across the wave; force them into SGPRs so the
  // gather/atomic addressing is saddr + lane*16.
  s = __builtin_amdgcn_readfirstlane(s);
  d = __builtin_amdgcn_readfirstlane(d);
  const float4 v = x4[(size_t)s * (D_FEAT / 4) + lane];
  float* o = out + (size_t)d * D_FEAT + lane * 4;
  unsafeAtomicAdd(o + 0, v.x);
  unsafeAtomicAdd(o + 1, v.y);
  unsafeAtomicAdd(o + 2, v.z);
  unsafeAtomicAdd(o + 3, v.w);
}

__global__ __launch_bounds__(TPB) void gnn_scatter_kernel(
    const float* __restrict__ x, const int* __restrict__ src_idx,
    const int* __restrict__ dst_idx, float* __restrict__ out,
    int n_edges, int n_chunks) {
  const int lane  = threadIdx.x & 31;
  const int wave  = __builtin_amdgcn_readfirstlane((int)(threadIdx.x >> 5));
  const int gwave = blockIdx.x * WPB + wave;
  const int nwaves = gridDim.x * WPB;
  const float4* x4 = reinterpret_cast<const float4*>(x);

#if USE_ASYNC_LDS
  // Per-wave double buffer of (src,dst) indices, filled by async global->LDS.
  __shared__ int lds_idx[WPB][2][2][CHUNK];

  auto issue = [&](int c, int b) {
    int e = c * CHUNK + lane;
    if (e >= n_edges) e = n_edges - 1;  // clamp (harmless duplicate fetch)
    __builtin_amdgcn_global_load_async_to_lds_b32(
        (gptr_i32)(src_idx + e),
        (lptr_i32)&lds_idx[wave][b][0][lane],
        0, 0);
    __builtin_amdgcn_global_load_async_to_lds_b32(
        (gptr_i32)(dst_idx + e),
        (lptr_i32)&lds_idx[wave][b][1][lane],
        0, 0);
  };

  int c = gwave, buf = 0;
  if (c < n_chunks) issue(c, 0);
  for (; c < n_chunks; c += nwaves) {
    const int cn = c + nwaves;
    if (cn < n_chunks) {
      issue(cn, buf ^ 1);                    // prefetch next chunk's indices
      __builtin_amdgcn_s_wait_asynccnt(2);   // current chunk's 2 ops are done
    } else {
      __builtin_amdgcn_s_wait_asynccnt(0);
    }
    asm volatile("" ::: "memory");           // keep LDS reads below the wait
    const int base = c * CHUNK;
    int lim = n_edges - base;
    if (lim > CHUNK) lim = CHUNK;
#pragma unroll 4
    for (int j = 0; j < lim; ++j)
      scatter_edge(x4, out, lds_idx[wave][buf][0][j], lds_idx[wave][buf][1][j],
                   lane);
    buf ^= 1;
  }
#else
  for (int c = gwave; c < n_chunks; c += nwaves) {
    const int base = c * CHUNK;
    int lim = n_edges - base;
    if (lim > CHUNK) lim = CHUNK;
    int pf = base + CHUNK * nwaves;
    if (pf > n_edges - 1) pf = n_edges - 1;
    __builtin_prefetch(src_idx + pf, 0, 0);  // global_prefetch_b8
    __builtin_prefetch(dst_idx + pf, 0, 0);
#pragma unroll 4
    for (int j = 0; j < lim; ++j)
      scatter_edge(x4, out, src_idx[base + j], dst_idx[base + j], lane);
  }
#endif
}

__global__ __launch_bounds__(TPB) void gnn_zero_kernel(float* __restrict__ out, int n4) {
  int i = blockIdx.x * blockDim.x + threadIdx.x;
  const int stride = gridDim.x * blockDim.x;
  const float4 z = make_float4(0.f, 0.f, 0.f, 0.f);
  for (; i < n4; i += stride)
    reinterpret_cast<float4*>(out)[i] = z;
}

extern "C" void kernel_launch(void* const* d_in, const int* in_sizes, int n_in,
                              void* d_out, int out_size, void* d_ws, size_t ws_size,
                              hipStream_t stream) {
  (void)n_in; (void)d_ws; (void)ws_size;
  const float* x  = (const float*)d_in[0];
  const int*   ei = (const int*)d_in[1];          // edge_index flat [2, E]
  const int n_edges = in_sizes[1] / 2;
  const int* src = ei;                            // row 0: source nodes
  const int* dst = ei + n_edges;                  // row 1: target nodes
  float* out = (float*)d_out;

  // Zero-init output (harness poisons d_out; atomics need zeros).
  const int n4 = out_size / 4;
  const int zb = (n4 + TPB - 1) / TPB;
  gnn_zero_kernel<<<zb, TPB, 0, stream>>>(out, n4);

  const int n_chunks = (n_edges + CHUNK - 1) / CHUNK;  // 10000
  int blocks = (n_chunks + WPB - 1) / WPB;
  if (blocks > 512) blocks = 512;                 // ~4096 waves, ~2.4 chunks each
  gnn_scatter_kernel<<<blocks, TPB, 0, stream>>>(x, src, dst, out, n_edges,
                                                 n_chunks);
}